// MoEGate_88141318849020
// MI455X (gfx1250) — compile-verified
//
#include <hip/hip_runtime.h>
#include <hip/hip_bf16.h>
#include <math.h>

// ---------------------------------------------------------------------------
// MoE gate for MI455X (gfx1250, wave32):
//   logits[T,E] = x[T,C] @ W[E,C]^T  via V_WMMA_F32_16X16X4_F32
//   Double-buffered software pipeline: the Tensor Data Mover DMAs chunk c+1's
//   X/W tiles into LDS (with hardware row padding -> bank-conflict-free
//   stride-36 layout) while the waves run WMMA on chunk c. One barrier/chunk.
// T = 16384 tokens, C = 4096, E = 64 experts, K = 8, B = 4 batches
// HBM-bound: 268 MB of x @ 23.3 TB/s ~ 11.5 us; W (1 MB) is L2-resident, so
// 256 small blocks (64 tokens each) maximize occupancy at no HBM cost.
// ---------------------------------------------------------------------------

typedef __attribute__((ext_vector_type(2))) float v2f;
typedef __attribute__((ext_vector_type(8))) float v8f;
typedef __attribute__((ext_vector_type(4))) unsigned int u32x4;
typedef __attribute__((ext_vector_type(4))) int          i32x4;
typedef __attribute__((ext_vector_type(8))) int          i32x8;

constexpr int Bc     = 4;
constexpr int Sc     = 4096;
constexpr int Cc     = 4096;
constexpr int Ec     = 64;
constexpr int Tc     = Bc * Sc;      // 16384 tokens
constexpr int TOPK   = 8;
constexpr int TILE_M = 64;           // tokens per block (4 waves x 16 rows)
constexpr int KC     = 32;           // K-chunk (dwords per tile row)
constexpr int LDSS   = 36;           // padded LDS row stride: 32 data + 4 pad
constexpr int NCHUNK = Cc / KC;      // 128
constexpr int BUFF   = Ec * LDSS;    // floats per tile buffer (64 rows)
constexpr int LSTR   = 65;           // logits overlay stride (bank = t+e)
constexpr float EPSc  = 1e-10f;
constexpr float ALPHA = 0.001f;

#if defined(__has_builtin)
#if __has_builtin(__builtin_amdgcn_tensor_load_to_lds) && \
    __has_builtin(__builtin_amdgcn_s_wait_tensorcnt)
#define USE_TDM 1
#endif
#endif
#ifndef USE_TDM
#define USE_TDM 0
#endif

#if USE_TDM && defined(__has_include)
#if __has_include(<hip/amd_detail/amd_gfx1250_TDM.h>)
#define TDM_6ARG 1     // amdgpu-toolchain / therock-10.0 headers: 6-arg builtin
#endif
#endif
#ifndef TDM_6ARG
#define TDM_6ARG 0
#endif

__device__ __forceinline__ float sigmoidf_(float v) {
    return 1.0f / (1.0f + __expf(-v));
}

#if USE_TDM
// D# group1 for a 2D tile: KC dwords/row, 64 rows, tensor row stride = Cc,
// LDS rows padded 32 data + 4 pad dwords (pad_interval=4 -> 32 DWORDs).
__device__ __forceinline__ i32x8 tdm_group1(int tensor_rows) {
    i32x8 g1;
    g1[0] = (int)((2u << 16)      // data_size = 4 bytes
                | (1u << 20)      // pad_enable
                | (4u << 22)      // pad_interval: every 32 DWORDs
                | (3u << 25));    // pad_amount: 4 DWORDs  -> LDS stride 36
    g1[1] = (int)(((unsigned)Cc & 0xFFFFu) << 16);            // tensor_dim0[15:0]
    g1[2] = (int)((((unsigned)Cc >> 16) & 0xFFFFu)
                | (((unsigned)tensor_rows & 0xFFFFu) << 16));  // dim0 hi | dim1 lo
    g1[3] = (int)((((unsigned)tensor_rows >> 16) & 0xFFFFu)
                | ((unsigned)KC << 16));                       // dim1 hi | tile_dim0
    g1[4] = Ec;                                                // tile_dim1 = 64 rows
    g1[5] = Cc;                                                // tensor_dim0_stride lo
    g1[6] = 0;
    g1[7] = 0;
    return g1;
}

__device__ __forceinline__ u32x4 tdm_group0(unsigned lds_addr,
                                            unsigned long long gaddr) {
    u32x4 g0;
    g0[0] = 1u;                                            // count=1, no gather
    g0[1] = lds_addr;                                      // LDS byte address
    g0[2] = (unsigned)gaddr;                               // global_addr[31:0]
    g0[3] = ((unsigned)(gaddr >> 32) & 0x01FFFFFFu)        // global_addr[56:32]
          | (2u << 30);                                    // type = 2 ("image")
    return g0;
}

__device__ __forceinline__ void tdm_issue(u32x4 g0, i32x8 g1) {
    i32x4 z4 = (i32x4){0, 0, 0, 0};
#if TDM_6ARG
    i32x8 z8 = (i32x8){0, 0, 0, 0, 0, 0, 0, 0};
    __builtin_amdgcn_tensor_load_to_lds(g0, g1, z4, z4, z8, 0);
#else
    __builtin_amdgcn_tensor_load_to_lds(g0, g1, z4, z4, 0);
#endif
}
#endif // USE_TDM

__global__ void __launch_bounds__(256)
moe_gate_zero_ws(float* ws) {
    int t = threadIdx.x;
    ws[t] = 0.0f;
    ws[t + 256] = 0.0f;
}

__global__ void __launch_bounds__(128)
moe_gate_main(const float* __restrict__ x,      // [T, C]
              const float* __restrict__ wgt,    // [E, C]
              const float* __restrict__ bias,   // [E]
              int*   __restrict__ out_idx,      // [T, 8]
              float* __restrict__ out_w,        // [T, 8]
              float* __restrict__ g_f,          // [B, E] selection counts
              float* __restrict__ g_p)          // [B, E] normalized-score sums
{
    // [X0][X1][W0][W1], each 64 x 36 floats. Logits (64 x 65) overlay X0+X1.
    __shared__ float smem[4 * BUFF];
    __shared__ float b_lds[Ec];
    __shared__ float p_sum[Ec];
    __shared__ int   f_cnt[Ec];

    float* X_lds = smem;                 // 2 buffers
    float* W_lds = smem + 2 * BUFF;      // 2 buffers

    const int tid   = threadIdx.x;
    const int lane  = tid & 31;
    const int wave  = tid >> 5;          // 0..3
    const int n     = lane & 15;
    const int hi    = lane >> 4;         // 0 or 1
    const int m0    = blockIdx.x * TILE_M;
    const int batch = m0 >> 12;          // 4096 tokens per batch

    if (tid < Ec) {
        b_lds[tid] = bias[tid];
        p_sum[tid] = 0.0f;
        f_cnt[tid] = 0;
    }

    v8f acc[4];
    #pragma unroll
    for (int g = 0; g < 4; ++g) acc[g] = (v8f){0,0,0,0,0,0,0,0};

#if USE_TDM
    // Loop-invariant descriptor pieces (uniform; wave 0 issues the DMAs).
    const unsigned lds_x = (unsigned)(size_t)(void*)X_lds;
    const unsigned lds_w = (unsigned)(size_t)(void*)W_lds;
    const i32x8 g1x = tdm_group1(/*tensor_rows=*/Tc);
    const i32x8 g1w = tdm_group1(/*tensor_rows=*/Ec);
    const unsigned long long gx0 = (unsigned long long)(size_t)(x + (size_t)m0 * Cc);
    const unsigned long long gw0 = (unsigned long long)(size_t)wgt;

    if (wave == 0) {                     // prologue: stage chunk 0 into buffer 0
        tdm_issue(tdm_group0(lds_x, gx0), g1x);
        tdm_issue(tdm_group0(lds_w, gw0), g1w);
        __builtin_amdgcn_s_wait_tensorcnt(0);
    }
    __syncthreads();
#endif

    for (int c = 0; c < NCHUNK; ++c) {
        const int buf = c & 1;
        const float* Xc = X_lds + buf * BUFF;
        const float* Wc = W_lds + buf * BUFF;

#if USE_TDM
        // Kick off chunk c+1 into the other buffer; DMA overlaps the WMMAs.
        if (wave == 0 && c + 1 < NCHUNK) {
            const unsigned boff = (unsigned)((buf ^ 1) * BUFF * 4);
            const unsigned long long koff = (unsigned long long)(c + 1) * KC * 4ull;
            tdm_issue(tdm_group0(lds_x + boff, gx0 + koff), g1x);
            tdm_issue(tdm_group0(lds_w + boff, gw0 + koff), g1w);
        }
#else
        {   // cooperative staging: 64 x 32 floats = 512 float4 per tile
            const int k0 = c * KC;
            #pragma unroll
            for (int i = 0; i < 4; ++i) {
                int idx = tid + i * 128;
                int row = idx >> 3;
                int c4  = (idx & 7) * 4;
                float4 vx = *(const float4*)(x + (size_t)(m0 + row) * Cc + k0 + c4);
                *(float4*)((float*)Xc + row * LDSS + c4) = vx;
                float4 vw = *(const float4*)(wgt + (size_t)row * Cc + k0 + c4);
                *(float4*)((float*)Wc + row * LDSS + c4) = vw;
            }
            if (c + 1 < NCHUNK) {
                int row = tid >> 1;
                __builtin_prefetch(x + (size_t)(m0 + row) * Cc + (k0 + KC) + (tid & 1) * 16, 0, 1);
            }
            __syncthreads();
        }
#endif

        // ---- 16x32 @ 32x64 per wave via WMMA f32 16x16x4 ----
        const float* xrow  = Xc + (wave * 16 + n) * LDSS;
        const float* wbase = Wc + n * LDSS;
        #pragma unroll
        for (int kk = 0; kk < KC; kk += 4) {
            const int ko = kk + 2 * hi;            // lane's K pair
            v2f a = *(const v2f*)(xrow + ko);      // A: row = lane&15
            #pragma unroll
            for (int g = 0; g < 4; ++g) {
                v2f b = *(const v2f*)(wbase + g * 16 * LDSS + ko); // B: col = lane&15
                acc[g] = __builtin_amdgcn_wmma_f32_16x16x4_f32(
                    false, a, false, b, (short)0, acc[g], false, false);
            }
        }

#if USE_TDM
        if (wave == 0) __builtin_amdgcn_s_wait_tensorcnt(0);
#endif
        __syncthreads();   // chunk consumed; next chunk's buffer ready (TDM path)
    }

    // ---- spill logits over the (dead) X buffers: logits[m][e], stride 65 ----
    float* logits_lds = X_lds;
    #pragma unroll
    for (int g = 0; g < 4; ++g) {
        #pragma unroll
        for (int r = 0; r < 8; ++r) {
            int m = wave * 16 + r + 8 * hi;
            logits_lds[m * LSTR + g * 16 + n] = acc[g][r];
        }
    }
    __syncthreads();

    // ---- per-token epilogue: sigmoid, p_i, biased top-8, weights ----
    if (tid < TILE_M) {
        const float* lrow = logits_lds + tid * LSTR;
        const int tok = m0 + tid;

        float denom = 0.0f;
        #pragma unroll 8
        for (int e = 0; e < Ec; ++e) denom += sigmoidf_(lrow[e]);
        const float inv_den = 1.0f / (denom + EPSc);
        #pragma unroll 8
        for (int e = 0; e < Ec; ++e)
            atomicAdd(&p_sum[e], sigmoidf_(lrow[e]) * inv_den);

        unsigned long long chosen = 0ull;
        float wsum = 0.0f;
        int   sel_i[TOPK];
        float sel_s[TOPK];
        #pragma unroll
        for (int k = 0; k < TOPK; ++k) {
            float best = -INFINITY;
            int   bi   = 0;
            for (int e = 0; e < Ec; ++e) {
                if ((chosen >> e) & 1ull) continue;
                float v = lrow[e] + b_lds[e];
                if (v > best) { best = v; bi = e; }   // strict > : lowest index on ties
            }
            chosen |= 1ull << bi;
            float s = sigmoidf_(lrow[bi]);
            sel_i[k] = bi;
            sel_s[k] = s;
            wsum += s;
            atomicAdd(&f_cnt[bi], 1);
        }
        const float inv_w = 1.0f / (wsum + EPSc);
        #pragma unroll
        for (int k = 0; k < TOPK; ++k) {
            out_idx[tok * TOPK + k] = sel_i[k];
            out_w[tok * TOPK + k]   = sel_s[k] * inv_w;
        }
    }
    __syncthreads();

    // ---- per-block -> per-batch partials ----
    if (tid < Ec) {
        atomicAdd(&g_f[batch * Ec + tid], (float)f_cnt[tid]);
        atomicAdd(&g_p[batch * Ec + tid], p_sum[tid]);
    }
}

__global__ void __launch_bounds__(256)
moe_gate_aux(const float* __restrict__ g_f,
             const float* __restrict__ g_p,
             float* __restrict__ out_loss)
{
    __shared__ float red[256];
    const int t = threadIdx.x;     // 256 = B*E entries
    float f_i = g_f[t] * (1.0f / (float)(TOPK * Sc));
    float p_i = g_p[t] * (1.0f / (float)Sc);
    red[t] = f_i * p_i;
    __syncthreads();
    #pragma unroll
    for (int s = 128; s > 0; s >>= 1) {
        if (t < s) red[t] += red[t + s];
        __syncthreads();
    }
    if (t == 0) out_loss[0] = ALPHA * red[0] * (1.0f / (float)Bc);
}

extern "C" void kernel_launch(void* const* d_in, const int* in_sizes, int n_in,
                              void* d_out, int out_size, void* d_ws, size_t ws_size,
                              hipStream_t stream) {
    (void)in_sizes; (void)n_in; (void)out_size; (void)ws_size;

    const float* x    = (const float*)d_in[0];   // [4,4096,4096]
    const float* wgt  = (const float*)d_in[1];   // [64,4096]
    const float* bias = (const float*)d_in[2];   // [64]

    // d_out layout: topk_idx [T*8] int32 ++ topk_weights [T*8] f32 ++ loss [1]
    int*   out_idx  = (int*)d_out;
    float* out_w    = ((float*)d_out) + (size_t)Tc * TOPK;
    float* out_loss = ((float*)d_out) + (size_t)2 * Tc * TOPK;

    float* g_f = (float*)d_ws;       // [B*E]
    float* g_p = g_f + 256;          // [B*E]

    moe_gate_zero_ws<<<1, 256, 0, stream>>>(g_f);
    moe_gate_main<<<Tc / TILE_M, 128, 0, stream>>>(x, wgt, bias, out_idx, out_w, g_f, g_p);
    moe_gate_aux<<<1, 256, 0, stream>>>(g_f, g_p, out_loss);
}